// InducedKernelGPR_77627238908269
// MI455X (gfx1250) — compile-verified
//
#include <hip/hip_runtime.h>

typedef float v2f __attribute__((ext_vector_type(2)));
typedef float v8f __attribute__((ext_vector_type(8)));

#define N_X      65536
#define N_IND    1024
#define N_DESC   64
#define CHUNK    128               // inducing rows staged in LDS per iteration
#define ROWPAD   68                // LDS row stride (floats): 64 + 4 -> conflict-free A fetch
#define THREADS  256
#define JTILES   2                 // 16-column B tiles kept in registers per wave
#define COLS_PER_WG (8 * 16 * JTILES)   // 8 waves * 32 columns = 256
#define LOG2E    1.44269504088896340736f

__global__ __launch_bounds__(THREADS)
void gpr_rbf_fused_wmma(const float* __restrict__ x,
                        const float* __restrict__ zin,
                        const float* __restrict__ alpha,
                        const float* __restrict__ ls,
                        float* __restrict__ out) {
  __shared__ float  s_zw[CHUNK * ROWPAD];   // weighted inducing chunk (padded rows)
  __shared__ float2 s_za[CHUNK];            // (||zw||^2 * log2e, alpha)
  __shared__ float  s_w[N_DESC];            // exp(-ls/2)

  const int tid  = threadIdx.x;
  const int lane = tid & 31;
  const int m    = lane & 15;               // column index within a 16-wide tile
  const int h    = lane >> 4;               // half-wave: K pair {0,1} vs {2,3}
  const int wave = tid >> 5;

  if (tid < N_DESC) s_w[tid] = __expf(-0.5f * ls[tid]);
  __syncthreads();

  // ---- per-wave B operands: JTILES x 16 columns, fp32, ISA 16x4-slice layout ----
  const int j0 = blockIdx.x * COLS_PER_WG + wave * (16 * JTILES);
  v2f   bt[JTILES][16];
  float xsqL[JTILES];                       // ||xw||^2 * log2e per tile
  #pragma unroll
  for (int t = 0; t < JTILES; ++t) {
    const float* xrow = x + (size_t)(j0 + t * 16 + m) * N_DESC + 2 * h;
    float xs_part = 0.f;
    #pragma unroll
    for (int s = 0; s < 16; ++s) {
      float2 xv = *(const float2*)(xrow + 4 * s);
      float w0 = s_w[4 * s + 2 * h];
      float w1 = s_w[4 * s + 2 * h + 1];
      v2f b; b.x = xv.x * w0; b.y = xv.y * w1;
      bt[t][s] = b;
      xs_part = fmaf(b.x, b.x, fmaf(b.y, b.y, xs_part));
    }
    xsqL[t] = (xs_part + __shfl_xor(xs_part, 16, 32)) * LOG2E;
  }

  float acc0[2] = {0.f, 0.f};               // independent fmac chains, tile 0
  float acc1[2] = {0.f, 0.f};               // independent fmac chains, tile 1

  for (int ic0 = 0; ic0 < N_IND; ic0 += CHUNK) {
    __syncthreads();  // previous chunk fully consumed before overwrite

    // ---- cooperative stage: each thread loads half a row (32 floats), weighted ----
    {
      const int r  = tid >> 1;
      const int hh = (tid & 1) * 32;
      const float* src = zin + (size_t)(ic0 + r) * N_DESC + hh;
      float* dst = &s_zw[r * ROWPAD + hh];
      #pragma unroll
      for (int q = 0; q < 32; q += 4) {
        float4 v = *(const float4*)(src + q);
        dst[q + 0] = v.x * s_w[hh + q + 0];
        dst[q + 1] = v.y * s_w[hh + q + 1];
        dst[q + 2] = v.z * s_w[hh + q + 2];
        dst[q + 3] = v.w * s_w[hh + q + 3];
      }
    }
    __syncthreads();

    if (tid < CHUNK) {
      const float* row = &s_zw[tid * ROWPAD];
      float ssum = 0.f;
      #pragma unroll
      for (int q = 0; q < N_DESC; ++q) ssum = fmaf(row[q], row[q], ssum);
      s_za[tid] = make_float2(ssum * LOG2E, alpha[ic0 + tid]);
    }
    __syncthreads();

    if (ic0 + CHUNK < N_IND)  // warm next chunk into caches (global_prefetch_b8)
      __builtin_prefetch(zin + (size_t)(ic0 + CHUNK) * N_DESC + tid * 32, 0, 3);

    // ---- 8 i-blocks: shared-A dual WMMA chains (K=64) + exp2 epilogue ----
    #pragma unroll
    for (int ib = 0; ib < CHUNK / 16; ++ib) {
      const float* arow = &s_zw[(ib * 16 + m) * ROWPAD + 2 * h];
      v8f c0 = {};
      v8f c1 = {};
      #pragma unroll
      for (int s = 0; s < 16; ++s) {
        v2f a = *(const v2f*)(arow + 4 * s);      // conflict-free ds_load_b64
        c0 = __builtin_amdgcn_wmma_f32_16x16x4_f32(
                 false, a, false, bt[0][s], (short)0, c0, false, false);
        c1 = __builtin_amdgcn_wmma_f32_16x16x4_f32(
                 false, a, false, bt[1][s], (short)0, c1, false, false);
      }
      float zx[8], zy[8];
      #pragma unroll
      for (int r = 0; r < 8; ++r) {               // broadcast LDS reads
        float2 zv = s_za[ib * 16 + 8 * h + r];
        zx[r] = zv.x; zy[r] = zv.y;
      }
      #pragma unroll
      for (int r = 0; r < 8; ++r) {
        // exp2(2*log2e*cross - log2e*(z_sq + x_sq)) == exp(-d)
        float e0 = exp2f(fmaf(c0[r], 2.f * LOG2E, -zx[r]) - xsqL[0]);
        float e1 = exp2f(fmaf(c1[r], 2.f * LOG2E, -zx[r]) - xsqL[1]);
        acc0[r & 1] = fmaf(zy[r], e0, acc0[r & 1]);
        acc1[r & 1] = fmaf(zy[r], e1, acc1[r & 1]);
      }
    }
  }

  // combine chains, then the two M-halves (lane l and l^16 share a column)
  float t0 = acc0[0] + acc0[1];
  float t1 = acc1[0] + acc1[1];
  t0 += __shfl_xor(t0, 16, 32);
  t1 += __shfl_xor(t1, 16, 32);
  if (h == 0) {
    out[j0 + m]      = t0;
    out[j0 + 16 + m] = t1;
  }
}

extern "C" void kernel_launch(void* const* d_in, const int* in_sizes, int n_in,
                              void* d_out, int out_size, void* d_ws, size_t ws_size,
                              hipStream_t stream) {
  (void)in_sizes; (void)n_in; (void)out_size; (void)d_ws; (void)ws_size;
  const float* x     = (const float*)d_in[0];
  const float* zin   = (const float*)d_in[1];
  const float* alpha = (const float*)d_in[2];
  const float* ls    = (const float*)d_in[3];
  float* out = (float*)d_out;

  dim3 grid(N_X / COLS_PER_WG);   // 256 workgroups
  gpr_rbf_fused_wmma<<<grid, THREADS, 0, stream>>>(x, zin, alpha, ls, out);
}